// DDRFMixer_26010321945264
// MI455X (gfx1250) — compile-verified
//
#include <hip/hip_runtime.h>

typedef float v2f __attribute__((ext_vector_type(2)));
typedef float v8f __attribute__((ext_vector_type(8)));

#define BB      4
#define TT      4096
#define DD      1024
#define NTAPS   7
#define TILE_T  128
#define NTHREADS 256

// out[b,t,d] = sum_n softmax_n( x[b,t,:] . W[n,:] ) * x[b, t - off_n, d]
__global__ __launch_bounds__(NTHREADS)
void ddrf_mixer_kernel(const float* __restrict__ x,
                       const float* __restrict__ W,
                       float* __restrict__ out)
{
    // WMMA B-operand layout: sW[(k>>1)*16 + n*2 + (k&1)] = W[n][k] for taps n<7.
    // Tap column 7 is never written (garbage) and taps 8..15 are served by
    // clamped duplicates -- C columns >= 7 are discarded, and WMMA column n
    // depends only on B column n, so no masking/zero-fill is needed.
    __shared__ float sW[(DD / 2) * 16];       // 32 KB
    __shared__ float sTap[TILE_T][8];         // logits, then softmax weights (4 KB)

    const int tid  = threadIdx.x;
    const int wgid = blockIdx.x;              // 0 .. B*T/TILE_T-1
    const int b    = wgid / (TT / TILE_T);
    const int t0   = (wgid % (TT / TILE_T)) * TILE_T;

    // ---------------- stage W into LDS (paired-K layout) ----------------
    for (int i = tid; i < NTAPS * DD; i += NTHREADS) {
        const int n = i >> 10;                // / DD
        const int k = i & (DD - 1);
        sW[(k >> 1) * 16 + n * 2 + (k & 1)] = W[i];
    }
    __syncthreads();

    // ---------------- Phase A: logits via V_WMMA_F32_16X16X4_F32 ----------------
    // 8 waves x one 16-time-row tile each; K = DD reduced in 4 independent chains.
    const int wave  = tid >> 5;
    const int lane  = tid & 31;
    const int mrow  = lane & 15;              // A row (time) / B col (tap)
    const int khalf = (lane >> 4) << 1;       // 0 (lanes 0-15) or 2 (lanes 16-31)
    const int bclamp = mrow & 7;              // clamped tap column (dup cols OK)

    const float* xrow =
        x + ((long)b * TT + (t0 + wave * 16 + mrow)) * DD;
    const v2f* sWp = (const v2f*)sW;          // pair index: (kb>>1)*8 + tap

    v8f acc[4];
    #pragma unroll
    for (int c = 0; c < 4; ++c)
        acc[c] = (v8f){0.f, 0.f, 0.f, 0.f, 0.f, 0.f, 0.f, 0.f};

    for (int k0 = 0; k0 < DD; k0 += 16) {
        #pragma unroll
        for (int c = 0; c < 4; ++c) {
            const int kb = k0 + c * 4 + khalf;          // even
            const v2f a  = *(const v2f*)(xrow + kb);     // A: x[t, kb..kb+1]
            const v2f bv = sWp[(kb >> 1) * 8 + bclamp];  // B: W[tap][kb..kb+1]
            acc[c] = __builtin_amdgcn_wmma_f32_16x16x4_f32(
                         false, a, false, bv, (short)0, acc[c], false, false);
        }
    }

    // C layout (16x16 f32): lanes 0-15 hold rows M=0..7 (VGPR r) of col N=lane,
    // lanes 16-31 hold rows M=8..15 of col N=lane-16.
    if (mrow < NTAPS) {
        const int rbase = wave * 16 + ((lane < 16) ? 0 : 8);
        #pragma unroll
        for (int r = 0; r < 8; ++r)
            sTap[rbase + r][mrow] = acc[0][r] + acc[1][r] + acc[2][r] + acc[3][r];
    }
    __syncthreads();

    // ---------------- Phase B: softmax over the 7 taps, per time row ----------------
    if (tid < TILE_T) {
        float l[NTAPS];
        float m = -3.4e38f;
        #pragma unroll
        for (int n = 0; n < NTAPS; ++n) { l[n] = sTap[tid][n]; m = fmaxf(m, l[n]); }
        float s = 0.0f;
        #pragma unroll
        for (int n = 0; n < NTAPS; ++n) { l[n] = __expf(l[n] - m); s += l[n]; }
        const float inv = 1.0f / s;
        #pragma unroll
        for (int n = 0; n < NTAPS; ++n) sTap[tid][n] = l[n] * inv;
    }
    __syncthreads();

    // ---------------- Phase C: softmax-weighted causal tap mix ----------------
    const int OFF[NTAPS] = {1, 2, 4, 8, 16, 32, 64};
    const float* xb = x + (long)b * TT * DD;
    float* ob = out + ((long)b * TT + t0) * DD;
    const int dcol = tid * 4;                  // one float4 of D per lane

    for (int tr = 0; tr < TILE_T; ++tr) {
        const int t = t0 + tr;
        float w[NTAPS];
        #pragma unroll
        for (int n = 0; n < NTAPS; ++n) w[n] = sTap[tr][n];   // LDS broadcast

        float4 a = make_float4(0.f, 0.f, 0.f, 0.f);
        #pragma unroll
        for (int n = 0; n < NTAPS; ++n) {
            const int ts = t - OFF[n];
            if (ts >= 0) {                     // wave-uniform causal pad
                const float4 xv = *(const float4*)(xb + (long)ts * DD + dcol);
                a.x += w[n] * xv.x;
                a.y += w[n] * xv.y;
                a.z += w[n] * xv.z;
                a.w += w[n] * xv.w;
            }
        }
        *(float4*)(ob + (long)tr * DD + dcol) = a;
    }
}

extern "C" void kernel_launch(void* const* d_in, const int* in_sizes, int n_in,
                              void* d_out, int out_size, void* d_ws, size_t ws_size,
                              hipStream_t stream)
{
    const float* x = (const float*)d_in[0];   // [B,T,D] f32
    const float* W = (const float*)d_in[1];   // [7,D]  f32
    float* out = (float*)d_out;               // [B,T,D] f32
    const int nblocks = BB * (TT / TILE_T);   // 128 workgroups
    ddrf_mixer_kernel<<<nblocks, NTHREADS, 0, stream>>>(x, W, out);
}